// ActiveProbCalc_83708912599352
// MI455X (gfx1250) — compile-verified
//
#include <hip/hip_runtime.h>
#include <hip/hip_bf16.h>
#include <math.h>

// ---------------------------------------------------------------------------
// Jagged attention log-softmax over template candidates (CDNA5 / gfx1250).
//
// One wave32 per segment (graph). Dot products computed with
// V_WMMA_F32_16X16X4_F32: A = 16 candidate template rows (16xK f32),
// B = graph embedding replicated across all 16 columns, so every output
// column of the 16x16 accumulator equals the 16 desired logits.
// D=128 -> 32 chained WMMAs per 16-candidate tile.
// Jagged softmax uses a streaming (online) logsumexp so no per-candidate
// logits buffer is needed.
// ---------------------------------------------------------------------------

typedef __attribute__((ext_vector_type(2))) float v2f;
typedef __attribute__((ext_vector_type(8))) float v8f;

#define EMB_D 128

__global__ __launch_bounds__(32)
void ActiveProbCalc_kernel(const float* __restrict__ graph_embed,  // [B, D]
                           const float* __restrict__ tpl,          // [V+1, D]
                           const int*   __restrict__ cand_idx,     // [T]
                           const int*   __restrict__ seg_ids,      // [T] sorted
                           const int*   __restrict__ target_pos,   // [B]
                           float*       __restrict__ out,          // [B]
                           int T)
{
    const int b    = (int)blockIdx.x;
    const int lane = (int)threadIdx.x;   // 0..31 (wave32)
    const int h    = lane >> 4;          // 0: lanes 0-15, 1: lanes 16-31
    const int m16  = lane & 15;          // row within A-fragment

    // --- segment bounds: lower_bound(seg_ids, b) / lower_bound(seg_ids, b+1)
    // (uniform across the wave -> scalar loads/branches)
    int lo = 0, hi = T;
    while (lo < hi) { int mid = (lo + hi) >> 1; if (seg_ids[mid] < b)     lo = mid + 1; else hi = mid; }
    const int start = lo;
    lo = start; hi = T;
    while (lo < hi) { int mid = (lo + hi) >> 1; if (seg_ids[mid] < b + 1) lo = mid + 1; else hi = mid; }
    const int end = lo;
    const int L   = end - start;         // >= 1 by construction

    const float* __restrict__ g = graph_embed + (size_t)b * EMB_D;

    // --- hoist the B-matrix fragments of g (replicated-column layout):
    // for WMMA step t, lanes 0-15 need g[4t..4t+1], lanes 16-31 need g[4t+2..4t+3]
    v2f bfrag[32];
#pragma unroll
    for (int t = 0; t < 32; ++t)
        bfrag[t] = *(const v2f*)(g + 4 * t + 2 * h);

    // --- streaming logsumexp over tiles of 16 candidates
    float runM = -INFINITY;
    float runS = 0.0f;

    for (int tile = 0; tile < L; tile += 16) {
        // A-fragment row for this lane: candidate (tile + m16), clamped at tail
        int co = tile + m16;
        int cc = start + (co < L ? co : (L - 1));
        const float* __restrict__ arow = tpl + (size_t)cand_idx[cc] * EMB_D;

        // prefetch next tile's row for this lane (clamped -> no divergence)
        int cn = tile + 16 + m16;
        int cp = start + (cn < L ? cn : (L - 1));
        __builtin_prefetch(tpl + (size_t)cand_idx[cp] * EMB_D + 2 * h, 0, 0);

        // 16x16 logits tile: D = A(16xK) * B(Kx16) + D, K chunks of 4
        v8f c = {};
#pragma unroll
        for (int t = 0; t < 32; ++t) {
            v2f a = *(const v2f*)(arow + 4 * t + 2 * h);
            c = __builtin_amdgcn_wmma_f32_16x16x4_f32(
                    /*neg_a=*/false, a, /*neg_b=*/false, bfrag[t],
                    /*c_mod=*/(short)0, c, /*reuse_a=*/false, /*reuse_b=*/false);
        }

        // C/D layout: element r of lane l holds candidate tile + r + 8*(l>=16)
        // (all columns identical since B's columns are identical).
        float vals[8];
        float tmax = -INFINITY;
#pragma unroll
        for (int r = 0; r < 8; ++r) {
            int o   = tile + r + 8 * h;
            float v = (o < L) ? c[r] : -INFINITY;   // mask jagged tail
            vals[r] = v;
            tmax = fmaxf(tmax, v);
        }
        // combine the two 16-lane halves (each half is internally uniform)
        tmax = fmaxf(tmax, __shfl_xor(tmax, 16, 32));

        float tsum = 0.0f;
#pragma unroll
        for (int r = 0; r < 8; ++r)
            tsum += expf(vals[r] - tmax);           // exp(-inf)=0 for masked
        tsum += __shfl_xor(tsum, 16, 32);

        // merge tile (tmax, tsum) into running (runM, runS)
        float newM = fmaxf(runM, tmax);
        runS = runS * expf(runM - newM) + tsum * expf(tmax - newM);
        runM = newM;
    }

    // --- target logit: coalesced lane-parallel dot (4 floats per lane)
    const int tp = target_pos[b];
    const float* __restrict__ trow = tpl + (size_t)cand_idx[tp] * EMB_D;
    float4 tv = *(const float4*)(trow + lane * 4);
    float4 gv = *(const float4*)(g    + lane * 4);
    float p = tv.x * gv.x + tv.y * gv.y + tv.z * gv.z + tv.w * gv.w;
#pragma unroll
    for (int off = 16; off >= 1; off >>= 1)
        p += __shfl_xor(p, off, 32);

    if (lane == 0)
        out[b] = p - runM - logf(runS);
}

extern "C" void kernel_launch(void* const* d_in, const int* in_sizes, int n_in,
                              void* d_out, int out_size, void* d_ws, size_t ws_size,
                              hipStream_t stream)
{
    const float* graph_embed = (const float*)d_in[0];   // [B, D] f32
    const float* tpl         = (const float*)d_in[1];   // [V+1, D] f32
    const int*   cand_idx    = (const int*)  d_in[2];   // [T] i32
    const int*   seg_ids     = (const int*)  d_in[3];   // [T] i32
    const int*   target_pos  = (const int*)  d_in[4];   // [B] i32
    float*       out         = (float*)d_out;           // [B] f32

    const int T = in_sizes[2];
    const int B = in_sizes[4];

    ActiveProbCalc_kernel<<<B, 32, 0, stream>>>(
        graph_embed, tpl, cand_idx, seg_ids, target_pos, out, T);
}